// DoubleSubstitutionHeadAutoRegressive_31585189495446
// MI455X (gfx1250) — compile-verified
//
#include <hip/hip_runtime.h>
#include <cmath>

// ---------------- constants from the reference ----------------
#define BATCH 4
#define EMB   256
#define HEAD  64
#define LEN2  8192
#define MIX2  2048
#define MIX1  8192
#define LEN1  (8 * MIX2)   // 16384
#define LEN0  (8 * MIX1)   // 65536
#define TOTV  (LEN2 + LEN1 + LEN0)  // 90112
#define OFF1  LEN2
#define OFF0  (LEN2 + LEN1)

typedef __attribute__((ext_vector_type(2))) float v2f;
typedef __attribute__((ext_vector_type(8))) float v8f;

__device__ __forceinline__ float gelu_f(float x) {
  return 0.5f * x * (1.0f + erff(x * 0.70710678118654752440f));
}

__device__ __forceinline__ v8f wmma_f32_16x16x4(v2f a, v2f b, v8f c) {
#if defined(__HIP_DEVICE_COMPILE__)
  // 8-arg pattern: (neg_a, A, neg_b, B, c_mod, C, reuse_a, reuse_b)
  return __builtin_amdgcn_wmma_f32_16x16x4_f32(false, a, false, b, (short)0, c,
                                               false, false);
#else
  (void)a; (void)b;
  return c;
#endif
}

// ---------------- generic WMMA GEMM ----------------
// D[M,N] = op(A)[M,K] @ W[K,N] + bias[col & 63] (+ addend[M,N])
// op = gelu if geluA. Requires M%64==0, N%32==0, K%32==0 (true for all calls).
#define TILE_M 64
#define TILE_N 32
#define TILE_K 32

__launch_bounds__(256)
__global__ void gemm_wmma(const float* __restrict__ A, const float* __restrict__ W,
                          const float* __restrict__ bias,
                          const float* __restrict__ addend,
                          float* __restrict__ D, int M, int N, int K, int geluA) {
  __shared__ float As[TILE_M * 33];
  __shared__ float Bs[TILE_K * 33];
  const int tid  = threadIdx.x;
  const int wid  = tid >> 5;
  const int lane = tid & 31;
  const int half = lane >> 4;     // 0: lanes 0-15, 1: lanes 16-31
  const int l16  = lane & 15;
  const int mt   = (wid & 3) * 16;   // wave's 16-row sub-tile within 64
  const int nt   = (wid >> 2) * 16;  // wave's 16-col sub-tile within 32
  const size_t m0 = (size_t)blockIdx.y * TILE_M;
  const size_t n0 = (size_t)blockIdx.x * TILE_N;

  v8f acc = {0.f, 0.f, 0.f, 0.f, 0.f, 0.f, 0.f, 0.f};

  for (int k0 = 0; k0 < K; k0 += TILE_K) {
    // stage gelu(A) tile 64x32 (coalesced: A is row-major, stride K)
#pragma unroll
    for (int i = 0; i < 8; ++i) {
      int e = tid + i * 256;
      int r = e >> 5, c = e & 31;
      float v = A[(m0 + (size_t)r) * (size_t)K + (size_t)(k0 + c)];
      if (geluA) v = gelu_f(v);
      As[r * 33 + c] = v;
    }
    // stage W tile 32x32 (row-major, stride N)
#pragma unroll
    for (int i = 0; i < 4; ++i) {
      int e = tid + i * 256;
      int r = e >> 5, c = e & 31;
      Bs[r * 33 + c] = W[(size_t)(k0 + r) * (size_t)N + n0 + (size_t)c];
    }
    __syncthreads();

#pragma unroll
    for (int kk = 0; kk < TILE_K; kk += 4) {
      const int ka = kk + (half ? 2 : 0);
      v2f a, b;
      a.x = As[(mt + l16) * 33 + ka];
      a.y = As[(mt + l16) * 33 + ka + 1];
      b.x = Bs[ka * 33 + nt + l16];
      b.y = Bs[(ka + 1) * 33 + nt + l16];
      acc = wmma_f32_16x16x4(a, b, acc);
    }
    __syncthreads();
  }

  const int col = (int)n0 + nt + l16;
  const float bv = bias ? bias[col & 63] : 0.0f;
#pragma unroll
  for (int r = 0; r < 8; ++r) {
    size_t row = m0 + (size_t)(mt + r + 8 * half);
    size_t off = row * (size_t)N + (size_t)col;
    float v = acc[r] + bv;
    if (addend) v += addend[off];
    D[off] = v;
  }
}

// ---------------- helper kernels ----------------
// causal-mask flatten: W(s,s,64,64) -> WF(s*64, s*64), WF[j*64+c][k*64+d] = (k>j)?W:0
__global__ void flatten_conv(const float* __restrict__ W, float* __restrict__ WF, int s) {
  const int K = s * 64;
  const size_t total = (size_t)K * K;
  for (size_t e = (size_t)blockIdx.x * blockDim.x + threadIdx.x; e < total;
       e += (size_t)gridDim.x * blockDim.x) {
    int rowi = (int)(e / K), coli = (int)(e % K);
    int j = rowi >> 6, c = rowi & 63;
    int k = coli >> 6, d = coli & 63;
    WF[e] = (k > j) ? W[(((size_t)(j * s + k) * 64 + c) << 6) + d] : 0.0f;
  }
}

// positions where value==2 (stable order) == argsort(where(mask,0,1))[:mix]
__global__ void build_idx(const int* __restrict__ value, int* __restrict__ idx,
                          int offset, int len, int mix) {
  if (threadIdx.x != 0) return;
  const int b = blockIdx.x;
  const int* v = value + (size_t)b * TOTV + offset;
  int cnt = 0;
  for (int t = 0; t < len && cnt < mix; ++t)
    if (v[t] == 2) idx[b * mix + cnt++] = t;
}

__global__ void embed_lookup(const int* __restrict__ value, const float* __restrict__ embT,
                             float* __restrict__ out) {
  size_t e = (size_t)blockIdx.x * blockDim.x + threadIdx.x;  // B*LEN0*64 threads
  int d = (int)(e & 63);
  size_t t = e >> 6;
  int b  = (int)(t / LEN0);
  int tt = (int)(t % LEN0);
  int tok = value[(size_t)b * TOTV + OFF0 + tt];
  out[e] = embT[(tok << 6) + d];
}

__global__ void zero_f32(float* __restrict__ p, size_t n) {
  size_t i = (size_t)blockIdx.x * blockDim.x + threadIdx.x;
  if (i < n) p[i] = 0.0f;
}

// dst[b, idx[b,i], :] = src[b, i*8+7, :]   (src: (B, 8*mix, 64), dst: (B, dstLen, 64))
__global__ void scatter_rows(const float* __restrict__ src, float* __restrict__ dst,
                             const int* __restrict__ idx, int mix, int dstLen) {
  size_t e = (size_t)blockIdx.x * blockDim.x + threadIdx.x;  // B*mix*64
  int d = (int)(e & 63);
  size_t r = e >> 6;
  int b = (int)(r / mix), i = (int)(r % mix);
  int di = idx[b * mix + i];
  dst[(((size_t)b * dstLen + di) << 6) + d] =
      src[(((size_t)b * (mix * 8) + (size_t)i * 8 + 7) << 6) + d];
}

// dst[b,i,:] = src[b, idx[b,i], :]   (src: (B, srcLen, 64), dst: (B, mix, 64))
__global__ void gather_rows(const float* __restrict__ src, float* __restrict__ dst,
                            const int* __restrict__ idx, int mix, int srcLen) {
  size_t e = (size_t)blockIdx.x * blockDim.x + threadIdx.x;  // B*mix*64
  int d = (int)(e & 63);
  size_t r = e >> 6;
  int b = (int)(r / mix), i = (int)(r % mix);
  int di = idx[b * mix + i];
  dst[(((size_t)b * mix + i) << 6) + d] =
      src[(((size_t)b * srcLen + di) << 6) + d];
}

// out[m, 0:3] = gelu(h[m,:]) @ l_W1(64,3) + l_b1
__global__ void final_proj(const float* __restrict__ h, const float* __restrict__ W,
                           const float* __restrict__ bias, float* __restrict__ out,
                           size_t Mrows) {
  size_t m = (size_t)blockIdx.x * blockDim.x + threadIdx.x;
  if (m >= Mrows) return;
  float a0 = bias[0], a1 = bias[1], a2 = bias[2];
  const float* hr = h + (m << 6);
#pragma unroll 8
  for (int c = 0; c < 64; ++c) {
    float g = gelu_f(hr[c]);
    a0 += g * W[c * 3 + 0];
    a1 += g * W[c * 3 + 1];
    a2 += g * W[c * 3 + 2];
  }
  out[m * 3 + 0] = a0;
  out[m * 3 + 1] = a1;
  out[m * 3 + 2] = a2;
}

// ---------------- launch ----------------
extern "C" void kernel_launch(void* const* d_in, const int* in_sizes, int n_in,
                              void* d_out, int out_size, void* d_ws, size_t ws_size,
                              hipStream_t stream) {
  (void)in_sizes; (void)n_in; (void)out_size; (void)ws_size;
  // inputs: setup_inputs() order, params dict flattened in insertion order
  const float* x     = (const float*)d_in[0];
  const int*   value = (const int*)d_in[1];
  // d_in[2]=depth, d_in[3]=pos unused
  const float* embT  = (const float*)d_in[4];
  const float* d2_W  = (const float*)d_in[5];
  const float* d2_b  = (const float*)d_in[6];
  const float* d2c_W = (const float*)d_in[7];
  const float* d2c_b = (const float*)d_in[8];
  const float* d1_W  = (const float*)d_in[9];
  const float* d1_b  = (const float*)d_in[10];
  const float* d1c_W = (const float*)d_in[11];
  const float* d1c_b = (const float*)d_in[12];
  const float* d0_W  = (const float*)d_in[13];
  const float* d0_b  = (const float*)d_in[14];
  const float* d0c_W = (const float*)d_in[15];
  const float* d0c_b = (const float*)d_in[16];
  const float* c2_W0 = (const float*)d_in[17];
  const float* c2_b0 = (const float*)d_in[18];
  const float* c2_W1 = (const float*)d_in[19];
  const float* c2_b1 = (const float*)d_in[20];
  const float* c1_W0 = (const float*)d_in[21];
  const float* c1_b0 = (const float*)d_in[22];
  const float* c1_W1 = (const float*)d_in[23];
  const float* c1_b1 = (const float*)d_in[24];
  const float* c0_W0 = (const float*)d_in[25];
  const float* c0_b0 = (const float*)d_in[26];
  const float* c0_W1 = (const float*)d_in[27];
  const float* c0_b1 = (const float*)d_in[28];
  const float* l_W0  = (const float*)d_in[29];
  const float* l_b0  = (const float*)d_in[30];
  const float* l_W1  = (const float*)d_in[31];
  const float* l_b1  = (const float*)d_in[32];
  float* out = (float*)d_out;

  // workspace layout (floats)
  float* ws = (float*)d_ws;
  size_t o = 0;
  float* wf00 = ws + o; o += 512 * 512;   // c0_W0 flat
  float* wf01 = ws + o; o += 512 * 512;   // c0_W1 flat
  float* wf10 = ws + o; o += 512 * 512;   // c1_W0 flat
  float* wf11 = ws + o; o += 512 * 512;   // c1_W1 flat
  float* wf20 = ws + o; o += 256 * 256;   // c2_W0 flat
  float* wf21 = ws + o; o += 256 * 256;   // c2_W1 flat
  float* emb0 = ws + o; o += (size_t)BATCH * LEN0 * 64;  // 16.7M
  float* emb1 = ws + o; o += (size_t)BATCH * LEN1 * 64;  // 4.2M
  float* emb2 = ws + o; o += (size_t)BATCH * LEN2 * 64;  // 2.1M
  float* tA   = ws + o; o += (size_t)BATCH * LEN0 * 64;  // 16.7M
  float* tB   = ws + o; o += (size_t)BATCH * LEN0 * 64;  // 16.7M
  int* idx1 = (int*)(ws + o); o += (size_t)BATCH * MIX1;
  int* idx2 = (int*)(ws + o); o += (size_t)BATCH * MIX2;

  auto gemm = [&](const float* A, const float* W, const float* bias,
                  const float* add, float* D, int M, int N, int K, int g) {
    dim3 grid(N / TILE_N, M / TILE_M);
    gemm_wmma<<<grid, 256, 0, stream>>>(A, W, bias, add, D, M, N, K, g);
  };

  // 0) flatten causal conv weights
  flatten_conv<<<1024, 256, 0, stream>>>(c0_W0, wf00, 8);
  flatten_conv<<<1024, 256, 0, stream>>>(c0_W1, wf01, 8);
  flatten_conv<<<1024, 256, 0, stream>>>(c1_W0, wf10, 8);
  flatten_conv<<<1024, 256, 0, stream>>>(c1_W1, wf11, 8);
  flatten_conv<<<256, 256, 0, stream>>>(c2_W0, wf20, 4);
  flatten_conv<<<256, 256, 0, stream>>>(c2_W1, wf21, 4);

  // 1) compaction index lists
  build_idx<<<BATCH, 32, 0, stream>>>(value, idx1, OFF1, LEN1, MIX1);
  build_idx<<<BATCH, 32, 0, stream>>>(value, idx2, 0, LEN2, MIX2);

  // 2) embedding + level-0 causal convs
  embed_lookup<<<(BATCH * (size_t)LEN0 * 64) / 256, 256, 0, stream>>>(value, embT, tA);
  gemm(tA, wf00, c0_b0, nullptr, tB, BATCH * LEN0 / 8, 512, 512, 0);
  gemm(tB, wf01, c0_b1, nullptr, emb0, BATCH * LEN0 / 8, 512, 512, 1);

  // 3) scatter to level-1, convs
  zero_f32<<<((size_t)BATCH * LEN1 * 64) / 256, 256, 0, stream>>>(emb1, (size_t)BATCH * LEN1 * 64);
  scatter_rows<<<((size_t)BATCH * MIX1 * 64) / 256, 256, 0, stream>>>(emb0, emb1, idx1, MIX1, LEN1);
  gemm(emb1, wf10, c1_b0, nullptr, tA, BATCH * LEN1 / 8, 512, 512, 1);
  gemm(tA, wf11, c1_b1, nullptr, emb1, BATCH * LEN1 / 8, 512, 512, 1);

  // 4) scatter to level-2, convs
  zero_f32<<<((size_t)BATCH * LEN2 * 64) / 256, 256, 0, stream>>>(emb2, (size_t)BATCH * LEN2 * 64);
  scatter_rows<<<((size_t)BATCH * MIX2 * 64) / 256, 256, 0, stream>>>(emb1, emb2, idx2, MIX2, LEN2);
  gemm(emb2, wf20, c2_b0, nullptr, tA, BATCH * LEN2 / 4, 256, 256, 1);
  gemm(tA, wf21, c2_b1, nullptr, emb2, BATCH * LEN2 / 4, 256, 256, 1);

  // 5) decode level-2: y2 = lin(gelu(deconv(gelu(x)))) + emb2
  gemm(x, d2_W, d2_b, nullptr, tA, BATCH * (LEN2 / 4), 256, 256, 1);        // (B,8192,64) as M=8192 rows x 256
  gemm(tA, d2c_W, d2c_b, emb2, tB, BATCH * LEN2, 64, 64, 1);                // y2

  // 6) gather -> level-1 decode
  gather_rows<<<((size_t)BATCH * MIX2 * 64) / 256, 256, 0, stream>>>(tB, tA, idx2, MIX2, LEN2);
  gemm(tA, d1_W, d1_b, nullptr, tB, BATCH * MIX2, 512, 64, 1);              // deconv -> (B,16384,64)
  gemm(tB, d1c_W, d1c_b, emb1, tA, BATCH * LEN1, 64, 64, 1);                // y1

  // 7) gather -> level-0 decode
  gather_rows<<<((size_t)BATCH * MIX1 * 64) / 256, 256, 0, stream>>>(tA, tB, idx1, MIX1, LEN1);
  gemm(tB, d0_W, d0_b, nullptr, tA, BATCH * MIX1, 512, 64, 1);              // deconv -> (B,65536,64)
  gemm(tA, d0c_W, d0c_b, emb0, tB, BATCH * LEN0, 64, 64, 1);                // y0

  // 8) head
  gemm(tB, l_W0, l_b0, nullptr, tA, BATCH * LEN0, 64, 64, 1);               // h
  final_proj<<<((size_t)BATCH * LEN0) / 256, 256, 0, stream>>>(tA, l_W1, l_b1, out,
                                                               (size_t)BATCH * LEN0);
}